// Channel_22196390986158
// MI455X (gfx1250) — compile-verified
//
#include <hip/hip_runtime.h>
#include <stdint.h>

// ---- CDNA5 (gfx1250) fused kernel: adaptive max-pool 3x3 -> 16x depthwise
// 3x3 "conv" (a 1x9x16 matmul via V_WMMA_F32_16X16X4_F32) -> sigmoid gate ->
// leaky scale of x. One workgroup per (b,c) plane; plane staged in LDS with
// GLOBAL_LOAD_ASYNC_TO_LDS_B128 so x is read from HBM exactly once.

typedef __attribute__((ext_vector_type(2))) float v2f;
typedef __attribute__((ext_vector_type(4))) float v4f;
typedef __attribute__((ext_vector_type(8))) float v8f;

// Async-LDS builtin wants pointee = 16-byte int vector, AS1 (global) src and
// AS3 (LDS) dst (clang prints AS1 as `__device__` in HIP diagnostics).
typedef int v4i __attribute__((vector_size(16)));
typedef __attribute__((address_space(1))) v4i gv4i;   // global v4i
typedef __attribute__((address_space(3))) v4i lv4i;   // LDS v4i

#define HW   9216   // 96*96
#define NT   256    // 8 wave32s
#define CCH  256    // channels

__device__ __forceinline__ float lrelu(float v) {
    // where(v>=0, v, 0.01v) == max(v, 0.01v) for all finite v
    return fmaxf(v, 0.01f * v);
}

// 16-byte global -> LDS async copy (per-lane), tracked by ASYNCcnt.
__device__ __forceinline__ void async_cp16(const float* g, float* l) {
#if __has_builtin(__builtin_amdgcn_global_load_async_to_lds_b128)
    __builtin_amdgcn_global_load_async_to_lds_b128(
        (gv4i*)(uintptr_t)g, (lv4i*)(uintptr_t)l, 0, 0);
#else
    asm volatile("global_load_async_to_lds_b128 %0, %1, off"
                 :: "v"((uint32_t)(uintptr_t)l), "v"((uint64_t)(uintptr_t)g)
                 : "memory");
#endif
}

__device__ __forceinline__ void wait_async0() {
#if __has_builtin(__builtin_amdgcn_s_wait_asynccnt)
    __builtin_amdgcn_s_wait_asynccnt(0);
#else
    asm volatile("s_wait_asynccnt 0" ::: "memory");
#endif
}

__global__ __launch_bounds__(NT)
void fused_pool_gate_scale(const float* __restrict__ x,
                           const float* __restrict__ w,
                           const float* __restrict__ bias,
                           float* __restrict__ out) {
    __shared__ __align__(16) float plane[HW];   // 36 KB
    __shared__ float red[9][8];                 // per-wave pool partials

    const int tid  = threadIdx.x;
    const int lane = tid & 31;
    const int wid  = tid >> 5;
    const int p    = blockIdx.x;        // p = b*256 + c
    const int c    = p & (CCH - 1);
    const float* xp = x   + (size_t)p * HW;
    float*       op = out + (size_t)p * HW;

    // ---- Stage 1: plane -> LDS via async b128 copies (9 per thread) ----
    {
        const float* g = xp + tid * 4;
        float*       l = &plane[tid * 4];
#pragma unroll
        for (int it = 0; it < 9; ++it) {
            async_cp16(g, l);
            g += NT * 4;
            l += NT * 4;
        }
    }
    wait_async0();        // this wave's async copies landed in LDS
    __syncthreads();      // all waves' copies visible

    // ---- Stage 2: 3x3 adaptive max pool (nine 32x32 block maxes) ----
#pragma unroll
    for (int cell = 0; cell < 9; ++cell) {
        const int py = cell / 3, px = cell % 3;
        float mm = -3.402823466e38f;
#pragma unroll
        for (int e = 0; e < 4; ++e) {           // 1024 elems / 256 threads
            const int idx = tid + e * NT;
            const int r = idx >> 5, cc = idx & 31;
            mm = fmaxf(mm, plane[(py * 32 + r) * 96 + (px * 32 + cc)]);
        }
#pragma unroll
        for (int mask = 16; mask >= 1; mask >>= 1)
            mm = fmaxf(mm, __shfl_xor(mm, mask, 32));
        if (lane == 0) red[cell][wid] = mm;
    }
    __syncthreads();

    float pooled[9];
#pragma unroll
    for (int cell = 0; cell < 9; ++cell) {      // broadcast reads, no conflicts
        float v = red[cell][0];
#pragma unroll
        for (int wv = 1; wv < 8; ++wv) v = fmaxf(v, red[cell][wv]);
        pooled[cell] = v;
    }

    // ---- Stage 3: conv[k] = sum_t pooled[t]*w[k,c,t]  (1x9 . 9x16) ----
    // Mapped onto 3 chained V_WMMA_F32_16X16X4_F32 (K padded 9->12 with 0s).
    // A (16x4 f32): lanes 0-15 carry K={0,1} in v[0],v[1]; lanes 16-31 K={2,3}.
    // A is replicated across M, so every D row equals conv[]; D VGPR0 in lane L
    // is conv[k = L&15]. All 8 waves run this identically => EXEC all 1s.
    const int   k  = lane & 15;
    const int   hi = lane >> 4;                       // half-wave selector
    const float* wp = w + ((size_t)(k * CCH + c)) * 9;
    float conv;
#if __has_builtin(__builtin_amdgcn_wmma_f32_16x16x4_f32)
    {
        v8f acc = {};
        v2f A, B;
        // chunk 0: K = {0,1} (lo half) / {2,3} (hi half)
        A.x = hi ? pooled[2] : pooled[0];
        A.y = hi ? pooled[3] : pooled[1];
        B.x = hi ? wp[2] : wp[0];
        B.y = hi ? wp[3] : wp[1];
        acc = __builtin_amdgcn_wmma_f32_16x16x4_f32(false, A, false, B,
                                                    (short)0, acc, false, false);
        // chunk 1: K = {4,5} / {6,7}
        A.x = hi ? pooled[6] : pooled[4];
        A.y = hi ? pooled[7] : pooled[5];
        B.x = hi ? wp[6] : wp[4];
        B.y = hi ? wp[7] : wp[5];
        acc = __builtin_amdgcn_wmma_f32_16x16x4_f32(false, A, false, B,
                                                    (short)0, acc, false, false);
        // chunk 2: K = {8,pad} / {pad,pad}
        A.x = hi ? 0.0f : pooled[8];
        A.y = 0.0f;
        B.x = hi ? 0.0f : wp[8];
        B.y = 0.0f;
        acc = __builtin_amdgcn_wmma_f32_16x16x4_f32(false, A, false, B,
                                                    (short)0, acc, false, false);
        conv = acc[0];
    }
#else
    {
        conv = 0.0f;
#pragma unroll
        for (int t = 0; t < 9; ++t) conv += pooled[t] * wp[t];
    }
#endif
    conv += bias[k * CCH + c];
    float s = 1.0f / (1.0f + __expf(-lrelu(conv)));
    // sum sigmoid over the 16 k's (each half-wave holds the full set)
#pragma unroll
    for (int mask = 1; mask <= 8; mask <<= 1) s += __shfl_xor(s, mask, 32);
    const float add = lrelu(s);   // gate > 0, but keep exact semantics

    // ---- Stage 4: out = leaky(add * x), streamed from LDS, NT stores ----
    const v4f* pl4 = (const v4f*)plane;
    v4f*       o4  = (v4f*)op;
#pragma unroll
    for (int it = 0; it < 9; ++it) {
        const int j = tid + it * NT;
        v4f v = pl4[j];
        v4f t = v * add;
        v4f r;
        r.x = fmaxf(t.x, 0.01f * t.x);
        r.y = fmaxf(t.y, 0.01f * t.y);
        r.z = fmaxf(t.z, 0.01f * t.z);
        r.w = fmaxf(t.w, 0.01f * t.w);
        __builtin_nontemporal_store(r, o4 + j);   // out never re-read: skip caches
    }
}

extern "C" void kernel_launch(void* const* d_in, const int* in_sizes, int n_in,
                              void* d_out, int out_size, void* d_ws, size_t ws_size,
                              hipStream_t stream) {
    (void)in_sizes; (void)n_in; (void)out_size; (void)d_ws; (void)ws_size;
    const float* x = (const float*)d_in[0];   // [32,256,96,96]
    const float* w = (const float*)d_in[1];   // [16,256,3,3]
    const float* b = (const float*)d_in[2];   // [16,256]
    float* out = (float*)d_out;               // [32,256,96,96]
    fused_pool_gate_scale<<<dim3(32 * 256), dim3(NT), 0, stream>>>(x, w, b, out);
}